// GraphClassificationModel_28157805593245
// MI455X (gfx1250) — compile-verified
//
#include <hip/hip_runtime.h>
#include <hip/hip_bf16.h>

typedef __attribute__((ext_vector_type(2))) float v2f;
typedef __attribute__((ext_vector_type(8))) float v8f;

#define TPB 256

// ---------------------------------------------------------------------------
// T[N,32] = X[N,K] @ W[K,32]   (f32 WMMA 16x16x4; one wave per 16-row tile,
// two 16-col accumulators). K in {64,32}, both multiples of 4. N may be any.
// ---------------------------------------------------------------------------
__global__ void gemm_xw_wmma(const float* __restrict__ X,
                             const float* __restrict__ W,
                             float* __restrict__ T, int N, int K) {
  const int wave = (blockIdx.x * blockDim.x + threadIdx.x) >> 5;
  const int lane = threadIdx.x & 31;
  const int row0 = wave * 16;
  if (row0 >= N) return;                    // wave-uniform exit

  const int mrow  = lane & 15;
  const int khalf = lane >> 4;              // 0: K pair {0,1}, 1: K pair {2,3}
  const int r     = row0 + mrow;
  const int rc    = (r < N) ? r : (N - 1);  // clamp loads (lane-level select only)

  v8f acc0 = {};
  v8f acc1 = {};
  for (int k0 = 0; k0 < K; k0 += 4) {
    const int ka = k0 + 2 * khalf;
    // A 16x4 f32: lanes 0-15 hold K={k0,k0+1}, lanes 16-31 hold K={k0+2,k0+3}
    v2f a = *(const v2f*)(X + (long)rc * K + ka);
    // B 4x16 f32: VGPR j, lane L -> K = k0 + 2*(L>>4) + j, N = L&15
    const float* wp = W + (long)ka * 32 + mrow;
    v2f b0 = { wp[0],  wp[32] };            // cols 0..15
    v2f b1 = { wp[16], wp[48] };            // cols 16..31
    acc0 = __builtin_amdgcn_wmma_f32_16x16x4_f32(false, a, false, b0,
                                                 (short)0, acc0, false, false);
    acc1 = __builtin_amdgcn_wmma_f32_16x16x4_f32(false, a, false, b1,
                                                 (short)0, acc1, false, false);
  }
  // D 16x16 f32: VGPR i, lanes 0-15 -> M=i, lanes 16-31 -> M=i+8; N = lane&15
#pragma unroll
  for (int i = 0; i < 8; ++i) {
    const int rr = row0 + i + 8 * khalf;
    if (rr < N) {
      T[(long)rr * 32 + mrow]      = acc0[i];
      T[(long)rr * 32 + 16 + mrow] = acc1[i];
    }
  }
}

// Fill H[total] with bias broadcast: H[i] = b[i & mask]  (mask = C-1, C pow2)
__global__ void bias_fill(float* __restrict__ H, const float* __restrict__ b,
                          int total, int mask) {
  int i = blockIdx.x * blockDim.x + threadIdx.x;
  if (i < total) H[i] = b[i & mask];
}

// Edge scatter, 32-wide messages: 8 threads/edge, float4 each.
__global__ void scatter_add32(const int* __restrict__ src,
                              const int* __restrict__ dst,
                              const float* __restrict__ T,
                              float* __restrict__ H, int E) {
  int tid = blockIdx.x * blockDim.x + threadIdx.x;
  int e = tid >> 3;
  if (e >= E) return;
  int g = (tid & 7) * 4;
  int s = src[e], d = dst[e];
  const float4 v = *(const float4*)(T + (long)s * 32 + g);
  float* o = H + (long)d * 32 + g;
  atomicAdd(o + 0, v.x);
  atomicAdd(o + 1, v.y);
  atomicAdd(o + 2, v.z);
  atomicAdd(o + 3, v.w);
}

// Edge scatter, 2-wide messages: 1 thread/edge.
__global__ void scatter_add2(const int* __restrict__ src,
                             const int* __restrict__ dst,
                             const float* __restrict__ T,
                             float* __restrict__ H, int E) {
  int e = blockIdx.x * blockDim.x + threadIdx.x;
  if (e >= E) return;
  int s = src[e], d = dst[e];
  v2f v = *(const v2f*)(T + (long)s * 2);
  atomicAdd(H + (long)d * 2 + 0, v.x);
  atomicAdd(H + (long)d * 2 + 1, v.y);
}

// T3[N,2] = H[N,32] @ W3[32,2]  (narrow output: plain VALU, W3 via scalar loads)
__global__ void transform2(const float* __restrict__ H,
                           const float* __restrict__ W3,
                           float* __restrict__ T3, int N) {
  int n = blockIdx.x * blockDim.x + threadIdx.x;
  if (n >= N) return;
  float o0 = 0.f, o1 = 0.f;
#pragma unroll
  for (int k = 0; k < 32; ++k) {
    float h = H[(long)n * 32 + k];
    o0 = fmaf(h, W3[2 * k + 0], o0);
    o1 = fmaf(h, W3[2 * k + 1], o1);
  }
  T3[(long)n * 2 + 0] = o0;
  T3[(long)n * 2 + 1] = o1;
}

// One block per graph. graph_ids sorted -> binary search [start,end), reduce
// the 64 original features, mean, dot with Wl, sigmoid. No atomics.
__global__ void pool_predict(const float* __restrict__ nf,
                             const int* __restrict__ gid,
                             const float* __restrict__ Wl,
                             const float* __restrict__ bl,
                             float* __restrict__ out, int N) {
  __shared__ float red[TPB];
  __shared__ float contrib[64];
  const int g = blockIdx.x;
  const int t = threadIdx.x;

  int lo = 0, hi = N;
  while (lo < hi) { int m = (lo + hi) >> 1; if (gid[m] < g) lo = m + 1; else hi = m; }
  const int start = lo;
  hi = N;
  while (lo < hi) { int m = (lo + hi) >> 1; if (gid[m] <= g) lo = m + 1; else hi = m; }
  const int end = lo;

  const int c = t & 63, sub = t >> 6;       // 4 node-strided partials per feat
  float acc = 0.f;
  for (int n = start + sub; n < end; n += 4) acc += nf[(long)n * 64 + c];
  red[t] = acc;
  __syncthreads();
  if (t < 64) {
    float s = red[t] + red[t + 64] + red[t + 128] + red[t + 192];
    float cnt = fmaxf((float)(end - start), 1.0f);
    contrib[t] = (s / cnt) * Wl[t];
  }
  __syncthreads();
  if (t == 0) {
    float tot = bl[0];
#pragma unroll
    for (int i = 0; i < 64; ++i) tot += contrib[i];
    out[g] = 1.0f / (1.0f + expf(-tot));
  }
}

extern "C" void kernel_launch(void* const* d_in, const int* in_sizes, int n_in,
                              void* d_out, int out_size, void* d_ws, size_t ws_size,
                              hipStream_t stream) {
  const float* node_feats = (const float*)d_in[0];
  // d_in[1] = edge_feats: unused by the reference forward pass
  const float* W1 = (const float*)d_in[2];
  const float* b1 = (const float*)d_in[3];
  const float* W2 = (const float*)d_in[4];
  const float* b2 = (const float*)d_in[5];
  const float* W3 = (const float*)d_in[6];
  const float* b3 = (const float*)d_in[7];
  const float* Wl = (const float*)d_in[8];
  const float* bl = (const float*)d_in[9];
  const int* src = (const int*)d_in[10];
  const int* dst = (const int*)d_in[11];
  const int* gid = (const int*)d_in[12];

  const int N = in_sizes[0] / 64;     // 100000 nodes
  const int E = in_sizes[10];         // 1600000 edges
  const int G = out_size - 2 * N;     // 64 graphs

  float* pred = (float*)d_out;        // [G]
  float* h3   = pred + G;             // [N,2], written in place

  float* bufA = (float*)d_ws;                 // [N,32] transformed messages
  float* bufB = bufA + (size_t)N * 32;        // [N,32] aggregated hidden

  const int gemmBlocks  = ((N + 15) / 16 + 7) / 8;        // 8 waves/block
  const int fill32Blk   = (N * 32 + TPB - 1) / TPB;
  const int scat32Blk   = (E * 8 + TPB - 1) / TPB;
  const int nodeBlk     = (N + TPB - 1) / TPB;
  const int fill2Blk    = (N * 2 + TPB - 1) / TPB;
  const int edgeBlk     = (E + TPB - 1) / TPB;

  // ---- Layer 1: t1 = X@W1 (WMMA), h1 = b1 + scatter(t1) ----
  gemm_xw_wmma<<<gemmBlocks, TPB, 0, stream>>>(node_feats, W1, bufA, N, 64);
  bias_fill<<<fill32Blk, TPB, 0, stream>>>(bufB, b1, N * 32, 31);
  scatter_add32<<<scat32Blk, TPB, 0, stream>>>(src, dst, bufA, bufB, E);

  // ---- Layer 2: t2 = h1@W2 (WMMA), h2 = b2 + scatter(t2) ----
  gemm_xw_wmma<<<gemmBlocks, TPB, 0, stream>>>(bufB, W2, bufA, N, 32);
  bias_fill<<<fill32Blk, TPB, 0, stream>>>(bufB, b2, N * 32, 31);
  scatter_add32<<<scat32Blk, TPB, 0, stream>>>(src, dst, bufA, bufB, E);

  // ---- Layer 3: t3 = h2@W3 (2-wide), h3 = b3 + scatter(t3) into d_out ----
  transform2<<<nodeBlk, TPB, 0, stream>>>(bufB, W3, bufA, N);
  bias_fill<<<fill2Blk, TPB, 0, stream>>>(h3, b3, N * 2, 1);
  scatter_add2<<<edgeBlk, TPB, 0, stream>>>(src, dst, bufA, h3, E);

  // ---- Graph readout: mean-pool original features -> linear -> sigmoid ----
  pool_predict<<<G, TPB, 0, stream>>>(node_feats, gid, Wl, bl, pred, N);
}